// MultiHeadAttentionLayer_67121748902425
// MI455X (gfx1250) — compile-verified
//
#include <hip/hip_runtime.h>
#include <hip/hip_bf16.h>

// ---------------------------------------------------------------------------
// MultiHeadAttention edge layer for MI455X (gfx1250, wave32).
//
//   Q = h @ WQ ; K = h @ WK ; V = h @ WV        (N x 128 @ 128 x 128, f32)
//   score[e] = K[src[e]] * Q[dst[e]] / sqrt(D)  (elementwise, 128 ch/edge)
//   wV = segsum(V[src]*score, dst) ; z = segsum(score, dst)
//   out = wV / (z + 1e-6)
//
// GEMM: V_WMMA_F32_16X16X4_F32 (native CDNA5 f32 WMMA), full f32 precision.
// GEMM is ~4 GFLOP -- negligible; the edge phase dominates and is
// L2-resident (Q/K/V/z ~82 MB < 192 MB L2): float4 gathers + f32 atomics.
// Edge lists are streamed once -> non-temporal loads keep them out of L2.
// ---------------------------------------------------------------------------

#define IN_DIM 128
#define HD     128           // H*D
#define INV_SQRT_D 0.25f     // 1/sqrt(16)

typedef float v2f __attribute__((ext_vector_type(2)));
typedef float v8f __attribute__((ext_vector_type(8)));

// ---------------------------------------------------------------- zero ------
__global__ void mha_zero2(float4* __restrict__ a, float4* __restrict__ b,
                          long n4) {
  long i = (long)blockIdx.x * blockDim.x + threadIdx.x;
  float4 zero = make_float4(0.f, 0.f, 0.f, 0.f);
  if (i < n4) { a[i] = zero; b[i] = zero; }
}

// ------------------------------------------------------- fused QKV GEMM -----
// grid = (mtiles, 3); block = 256 (8 waves).  blockIdx.y selects {Q,K,V};
// wave w computes the 16x16 tile (mtile, ntile=w).  All 8 waves share the
// same 16 rows of h -> A loads hit L1.
__global__ void mha_qkv_wmma(const float* __restrict__ h,
                             const float* __restrict__ WQ,
                             const float* __restrict__ WK,
                             const float* __restrict__ WV,
                             float* __restrict__ Q,
                             float* __restrict__ K,
                             float* __restrict__ V,
                             int nNodes) {
  const int lane  = threadIdx.x & 31;
  const int ntile = threadIdx.x >> 5;       // 0..7
  const int mtile = blockIdx.x;
  const int which = blockIdx.y;             // 0=Q 1=K 2=V

  if (mtile * 16 >= nNodes) return;         // uniform; EXEC all-1 inside

  const float* __restrict__ W = (which == 0) ? WQ : (which == 1) ? WK : WV;
  float*       __restrict__ O = (which == 0) ? Q  : (which == 1) ? K  : V;

  const int halfM = lane & 15;              // A row in tile / B,C,D column
  const int kk    = (lane >> 4) * 2;        // K sub-offset: 0 or 2
  const int row   = mtile * 16 + halfM;
  const int col   = ntile * 16 + halfM;

  v8f acc = {};
  const float* __restrict__ arow = h + (long)row * IN_DIM + kk;
  const float* __restrict__ bcol = W + (long)kk * HD + col;

  #pragma unroll
  for (int k = 0; k < IN_DIM; k += 4) {
    // A fragment (16x4 f32): lane -> A[row][k+kk], A[row][k+kk+1]
    v2f a;
    a.x = arow[k];
    a.y = arow[k + 1];
    // B fragment (4x16 f32): lane -> W[k+kk][col], W[k+kk+1][col]
    v2f b;
    b.x = bcol[(long)k * HD];
    b.y = bcol[(long)(k + 1) * HD];
    acc = __builtin_amdgcn_wmma_f32_16x16x4_f32(
        /*neg_a=*/false, a, /*neg_b=*/false, b,
        /*c_mod=*/(short)0, acc, /*reuse_a=*/false, /*reuse_b=*/false);
  }

  // C/D layout: VGPR v, lanes 0-15 -> M=v, lanes 16-31 -> M=v+8; N = lane&15
  const int crow = mtile * 16 + (lane >> 4) * 8;
  #pragma unroll
  for (int v = 0; v < 8; ++v) {
    O[(long)(crow + v) * HD + col] = acc[v];
  }
}

// ------------------------------------------------------------ edge phase ----
// One wave per edge; lane owns 4 contiguous channels of the 128.
__global__ void mha_edge(const int* __restrict__ src,
                         const int* __restrict__ dst,
                         const float* __restrict__ Q,
                         const float* __restrict__ K,
                         const float* __restrict__ V,
                         float* __restrict__ wV,
                         float* __restrict__ z,
                         int nEdges) {
  const int lane = threadIdx.x & 31;
  const long e = (long)blockIdx.x * (blockDim.x >> 5) + (threadIdx.x >> 5);
  if (e >= nEdges) return;

  // edge lists are streamed exactly once -> keep them out of L2 (NT loads)
  const int s = __builtin_nontemporal_load(src + e);
  const int d = __builtin_nontemporal_load(dst + e);
  const int c = lane * 4;

  const float4 kv = *(const float4*)(K + (long)s * HD + c);
  const float4 qv = *(const float4*)(Q + (long)d * HD + c);
  const float4 vv = *(const float4*)(V + (long)s * HD + c);

  const float s0 = kv.x * qv.x * INV_SQRT_D;
  const float s1 = kv.y * qv.y * INV_SQRT_D;
  const float s2 = kv.z * qv.z * INV_SQRT_D;
  const float s3 = kv.w * qv.w * INV_SQRT_D;

  float* wp = wV + (long)d * HD + c;
  float* zp = z  + (long)d * HD + c;
  atomicAdd(wp + 0, vv.x * s0);
  atomicAdd(wp + 1, vv.y * s1);
  atomicAdd(wp + 2, vv.z * s2);
  atomicAdd(wp + 3, vv.w * s3);
  atomicAdd(zp + 0, s0);
  atomicAdd(zp + 1, s1);
  atomicAdd(zp + 2, s2);
  atomicAdd(zp + 3, s3);
}

// -------------------------------------------------------------- finalize ----
__global__ void mha_finalize(float4* __restrict__ out,
                             const float4* __restrict__ z, long n4) {
  long i = (long)blockIdx.x * blockDim.x + threadIdx.x;
  if (i < n4) {
    float4 o = out[i];
    float4 zz = z[i];
    o.x = o.x / (zz.x + 1e-6f);
    o.y = o.y / (zz.y + 1e-6f);
    o.z = o.z / (zz.z + 1e-6f);
    o.w = o.w / (zz.w + 1e-6f);
    out[i] = o;
  }
}

// ---------------------------------------------------------------------------
extern "C" void kernel_launch(void* const* d_in, const int* in_sizes, int n_in,
                              void* d_out, int out_size, void* d_ws, size_t ws_size,
                              hipStream_t stream) {
  const float* h   = (const float*)d_in[0];
  const int*   src = (const int*)  d_in[1];
  const int*   dst = (const int*)  d_in[2];
  const float* WQ  = (const float*)d_in[3];
  const float* WK  = (const float*)d_in[4];
  const float* WV  = (const float*)d_in[5];

  const int N = in_sizes[0] / IN_DIM;   // 40000
  const int E = in_sizes[1];            // 640000
  const long NC = (long)N * HD;         // per-matrix float count

  float* Q = (float*)d_ws;
  float* K = Q + NC;
  float* V = K + NC;
  float* z = V + NC;                    // total ws: 4*NC floats (~82 MB)
  float* out = (float*)d_out;           // accumulates wV, then divided in-place

  // 1) zero accumulators (atomics accumulate; must reset every call)
  {
    const long n4 = NC / 4;
    const int blk = 256;
    const long grid = (n4 + blk - 1) / blk;
    mha_zero2<<<(int)grid, blk, 0, stream>>>((float4*)out, (float4*)z, n4);
  }

  // 2) fused Q/K/V GEMM via f32 WMMA: wave = one 16x16 tile
  {
    const int mtiles = (N + 15) / 16;   // 2500
    dim3 grid(mtiles, 3, 1);
    mha_qkv_wmma<<<grid, 256, 0, stream>>>(h, WQ, WK, WV, Q, K, V, N);
  }

  // 3) edge gather + atomic scatter (L2-resident)
  {
    const int blk = 256;                // 8 edges/block
    const int wavesPerBlock = blk / 32;
    const long grid = ((long)E + wavesPerBlock - 1) / wavesPerBlock;
    mha_edge<<<(int)grid, blk, 0, stream>>>(src, dst, Q, K, V, out, z, E);
  }

  // 4) out = wV / (z + 1e-6)
  {
    const long n4 = NC / 4;
    const int blk = 256;
    const long grid = (n4 + blk - 1) / blk;
    mha_finalize<<<(int)grid, blk, 0, stream>>>((float4*)out, (const float4*)z, n4);
  }
}